// GraphCNN_2808908611935
// MI455X (gfx1250) — compile-verified
//
#include <hip/hip_runtime.h>

#define NN 4096

typedef __attribute__((ext_vector_type(2))) float v2f;
typedef __attribute__((ext_vector_type(4))) float v4f;
typedef __attribute__((ext_vector_type(8))) float v8f;

// ---------------------------------------------------------------------------
// Tiny 3-layer MLP (8 -> 16 -> 8 -> 8), leaky_relu(0.01), one thread per row.
// mode 0: plain [NN,8] output (state / final output layout)
// mode 1: Q-packed B layout for WMMA: Q[row/4][16][4] floats, i.e.
//         Q[(row>>2)*64 + col*4 + (row&3)], cols 8..15 zero-filled.
// ---------------------------------------------------------------------------
__global__ void __launch_bounds__(256)
mlp3_kernel(const float* __restrict__ in,
            const float* __restrict__ W1, const float* __restrict__ b1,
            const float* __restrict__ W2, const float* __restrict__ b2,
            const float* __restrict__ W3, const float* __restrict__ b3,
            float* __restrict__ out, int mode)
{
    int row = blockIdx.x * 256 + threadIdx.x;
    if (row >= NN) return;

    float x[8], h1[16], h2[8];
#pragma unroll
    for (int i = 0; i < 8; ++i) x[i] = in[(size_t)row * 8 + i];

#pragma unroll
    for (int j = 0; j < 16; ++j) {
        float s = b1[j];
#pragma unroll
        for (int i = 0; i < 8; ++i) s = fmaf(x[i], W1[i * 16 + j], s);
        h1[j] = s > 0.f ? s : 0.01f * s;
    }
#pragma unroll
    for (int j = 0; j < 8; ++j) {
        float s = b2[j];
#pragma unroll
        for (int i = 0; i < 16; ++i) s = fmaf(h1[i], W2[i * 8 + j], s);
        h2[j] = s > 0.f ? s : 0.01f * s;
    }
    if (mode == 0) {
#pragma unroll
        for (int j = 0; j < 8; ++j) {
            float s = b3[j];
#pragma unroll
            for (int i = 0; i < 8; ++i) s = fmaf(h2[i], W3[i * 8 + j], s);
            out[(size_t)row * 8 + j] = s > 0.f ? s : 0.01f * s;
        }
    } else {
        float* q = out + (size_t)(row >> 2) * 64 + (row & 3);
#pragma unroll
        for (int j = 0; j < 8; ++j) {
            float s = b3[j];
#pragma unroll
            for (int i = 0; i < 8; ++i) s = fmaf(h2[i], W3[i * 8 + j], s);
            q[j * 4] = s > 0.f ? s : 0.01f * s;
        }
#pragma unroll
        for (int j = 8; j < 16; ++j) q[j * 4] = 0.f;   // zero-pad cols 8..15
    }
}

// ---------------------------------------------------------------------------
// Dout[NN,8] = A[NN,NN] @ B[NN,0:8] + Cin[NN,8]
//   (B supplied Q-packed: Q[g][16][4], g = K/4, cols 8..15 zero)
//
// One block per 16-row tile (256 blocks), 16 waves split K=4096 into 256-wide
// slices. Per loop iteration (K-step 8): one b128 NT load of A, one b128 load
// of Q, two V_WMMA_F32_16X16X4_F32 — 1 vmem instruction per WMMA.
//
// K-position mapping per WMMA (ISA f32 layouts, lo = lanes 0-15, hi = 16-31):
//   A: VGPR0/1 @ lo = K-pos 0/1, @ hi = K-pos 2/3 ; same for B rows.
// WMMA#1 covers K {k,k+1} (lo) ∪ {k+4,k+5} (hi);
// WMMA#2 covers K {k+2,k+3} (lo) ∪ {k+6,k+7} (hi);  sum over K is commutative.
// ---------------------------------------------------------------------------
__global__ void __launch_bounds__(512)
spmm_wmma_kernel(const float* __restrict__ A,    // [NN, NN] rem slice
                 const float* __restrict__ Q,    // [NN/4][16][4] packed agg
                 const float* __restrict__ Cin,  // [NN, 8]  previous state
                 float* __restrict__ Dout)       // [NN, 8]  next state
{
    __shared__ float red[16][256];

    const int tid  = threadIdx.x;
    const int wave = tid >> 5;
    const int lane = tid & 31;
    const int i0   = blockIdx.x << 4;      // first row of this tile

    const int m   = lane & 15;             // A row / B col within fragment
    const int khi = lane >> 4;             // 0 lo-half, 1 hi-half

    const int KW = NN / 16;                // 256 K-values per wave
    const int k0 = wave * KW;

    const float* __restrict__ arow = A + (size_t)(i0 + m) * NN + khi * 4;
    const float* __restrict__ qcol = Q + khi * 64 + m * 4;

    v8f acc = {};
    for (int k = k0; k < k0 + KW; k += 8) {
        v4f a = __builtin_nontemporal_load((const v4f*)(arow + k));
        v4f b = *(const v4f*)(qcol + (size_t)(k >> 2) * 64);
        v2f a0 = {a.x, a.y}, a1 = {a.z, a.w};
        v2f b0 = {b.x, b.y}, b1 = {b.z, b.w};
        acc = __builtin_amdgcn_wmma_f32_16x16x4_f32(
                  false, a0, false, b0, (short)0, acc, false, false);
        acc = __builtin_amdgcn_wmma_f32_16x16x4_f32(
                  false, a1, false, b1, (short)0, acc, false, false);
    }

    // Spill this wave's 16x16 accumulator tile to LDS.
#pragma unroll
    for (int r = 0; r < 8; ++r) red[wave][r * 32 + lane] = acc[r];
    __syncthreads();

    // Reduce 16 partial tiles; element (mm,n) lives at VGPR mm&7,
    // lane n + 16*(mm>=8). Only the 16x8 useful sub-tile is written.
    if (tid < 128) {
        const int mm = tid >> 3;                    // 0..15 row in tile
        const int n  = tid & 7;                     // 0..7  output feature
        const int r  = mm & 7;
        const int l  = ((mm >> 3) << 4) + n;
        float s = 0.f;
#pragma unroll
        for (int w = 0; w < 16; ++w) s += red[w][r * 32 + l];
        const size_t o = (size_t)(i0 + mm) * 8 + n;
        Dout[o] = s + Cin[o];
    }
}

// ---------------------------------------------------------------------------
// Host-side sequencing (graph-capture safe: only kernel launches on stream).
// ---------------------------------------------------------------------------
extern "C" void kernel_launch(void* const* d_in, const int* in_sizes, int n_in,
                              void* d_out, int out_size, void* d_ws, size_t ws_size,
                              hipStream_t stream) {
    const float* x   = (const float*)d_in[0];
    const float* rem = (const float*)d_in[1];
    const float* pW1 = (const float*)d_in[2];
    const float* pb1 = (const float*)d_in[3];
    const float* pW2 = (const float*)d_in[4];
    const float* pb2 = (const float*)d_in[5];
    const float* pW3 = (const float*)d_in[6];
    const float* pb3 = (const float*)d_in[7];
    const float* aW1 = (const float*)d_in[8];
    const float* ab1 = (const float*)d_in[9];
    const float* aW2 = (const float*)d_in[10];
    const float* ab2 = (const float*)d_in[11];
    const float* aW3 = (const float*)d_in[12];
    const float* ab3 = (const float*)d_in[13];

    float* out = (float*)d_out;

    float* s0  = (float*)d_ws;              // [NN, 8] state ping
    float* s1  = s0 + (size_t)NN * 8;       // [NN, 8] state pong
    float* agg = s1 + (size_t)NN * 8;       // [NN/4][16][4] packed agg

    // out0 = mlp_p(x)
    mlp3_kernel<<<NN / 256, 256, 0, stream>>>(x, pW1, pb1, pW2, pb2, pW3, pb3,
                                              s0, 0);

    const float* state = s0;
    for (int d = 0; d < 8; ++d) {
        // agg = mlp_a(state), Q-packed with zero cols 8..15
        mlp3_kernel<<<NN / 256, 256, 0, stream>>>(state, aW1, ab1, aW2, ab2,
                                                  aW3, ab3, agg, 1);
        float* dst = (d == 7) ? out : ((d & 1) ? s0 : s1);
        spmm_wmma_kernel<<<NN / 16, 512, 0, stream>>>(
            rem + (size_t)d * NN * NN, agg, state, dst);
        state = dst;
    }
}